// Decoder_12807592477243
// MI455X (gfx1250) — compile-verified
//
#include <hip/hip_runtime.h>
#include <hip/hip_bf16.h>
#include <math.h>

// ---- problem dims ----
#define VOCAB 50257
#define EMB   256
#define ENC2  1024
#define DEC   512
#define SRC   2048
#define BATCH 64

typedef __attribute__((ext_vector_type(2))) float v2f;
typedef __attribute__((ext_vector_type(8))) float v8f;
typedef __attribute__((ext_vector_type(4))) unsigned int v4u;
typedef __attribute__((ext_vector_type(8))) int v8i;
typedef __attribute__((ext_vector_type(4))) int v4i;

#if defined(__gfx1250__) && __has_builtin(__builtin_amdgcn_tensor_load_to_lds)
#define HAVE_TDM 1
#else
#define HAVE_TDM 0
#endif

// d_out layout (floats): prediction[64][50257] | h_new[64][512] | alignment[64][2048]
#define PRED_OFF  0
#define HNEW_OFF  ((size_t)BATCH * VOCAB)
#define ALIGN_OFF (HNEW_OFF + (size_t)BATCH * DEC)

// workspace layout (floats). wWT stored PAIR-INTERLEAVED: [k/2][n][2]
#define WS_WWT   0                                   // [1024/2][512][2]
#define WS_HPROJ (WS_WWT + (size_t)ENC2 * DEC)       // [64][512]
#define WS_EMBD  (WS_HPROJ + (size_t)BATCH * DEC)    // [64][256]
#define WS_ATT   (WS_EMBD + (size_t)BATCH * EMB)     // [64][2048]
#define WS_CTX   (WS_ATT + (size_t)BATCH * SRC)      // [64][1024]
#define WS_FEAT  (WS_CTX + (size_t)BATCH * ENC2)     // [64][1792]

// ---------------------------------------------------------------------------
// Tensor Data Mover: 2D tile load (global -> LDS), D# per CDNA5 ISA ch.8.
// data_size=4B. Optional LDS padding: pad_amount dwords (code) after every
// 2^(code+1) dwords. Issued wave-uniform; tracked by TENSORcnt (in-order).
// Toolchain uses the 6-arg builtin: (v4u, v8i, v4i, v4i, v8i, i32 cpol).
__device__ __forceinline__ void tdm_load_2d(unsigned lds_addr, const void* gptr,
                                            unsigned tile_d0, unsigned tile_d1,
                                            unsigned tensor_d0, unsigned tensor_d1,
                                            unsigned long long stride0_elems,
                                            unsigned pad_enable,
                                            unsigned pad_interval_code,
                                            unsigned pad_amount_code) {
#if HAVE_TDM
  unsigned long long ga = (unsigned long long)(uintptr_t)gptr;
  v4u g0;
  g0.x = 1u;                                                   // count=1 (valid)
  g0.y = lds_addr;                                             // LDS byte addr
  g0.z = (unsigned)(ga & 0xFFFFFFFFull);                       // global_addr lo
  g0.w = (unsigned)((ga >> 32) & 0x1FFFFFFull) | (2u << 30);   // addr hi | type=2
  unsigned w0 = (2u << 16);                                    // data_size = 4B
  if (pad_enable)
    w0 |= (1u << 20) | ((pad_interval_code & 7u) << 22) | ((pad_amount_code & 127u) << 25);
  v8i g1;
  g1[0] = (int)w0;
  g1[1] = (int)((tensor_d0 & 0xFFFFu) << 16);                  // dim0[15:0] @63:48
  g1[2] = (int)((tensor_d0 >> 16) | ((tensor_d1 & 0xFFFFu) << 16));
  g1[3] = (int)((tensor_d1 >> 16) | ((tile_d0 & 0xFFFFu) << 16));
  g1[4] = (int)(tile_d1 & 0xFFFFu);                            // tile_dim1; tile_dim2=0
  g1[5] = (int)(unsigned)(stride0_elems & 0xFFFFFFFFull);      // dim0 stride lo
  g1[6] = (int)((unsigned)((stride0_elems >> 32) & 0xFFFFull));// stride hi
  g1[7] = 0;
  v4i zero4 = {0, 0, 0, 0};
  v8i zero8 = {0, 0, 0, 0, 0, 0, 0, 0};
  __builtin_amdgcn_tensor_load_to_lds(g0, g1, zero4, zero4, zero8, 0);
#else
  (void)lds_addr; (void)gptr; (void)tile_d0; (void)tile_d1; (void)tensor_d0;
  (void)tensor_d1; (void)stride0_elems; (void)pad_enable; (void)pad_interval_code;
  (void)pad_amount_code;
#endif
}

// ---------------------------------------------------------------------------
// 1) gather embeddings
__global__ void k_embed(const int* __restrict__ input, const float* __restrict__ emb,
                        float* __restrict__ embedded) {
  int idx = blockIdx.x * blockDim.x + threadIdx.x;   // 64*256
  int b = idx >> 8, e = idx & 255;
  embedded[idx] = emb[(size_t)input[b] * EMB + e];
}

// 2) h_proj[b][d] = wb[d] + sum_{f<512} h[b][f] * wW[d][f]
__global__ void k_hproj(const float* __restrict__ h, const float* __restrict__ wW,
                        const float* __restrict__ wb, float* __restrict__ hproj) {
  int idx = blockIdx.x * blockDim.x + threadIdx.x;   // 64*512
  int b = idx / DEC, d = idx % DEC;
  const float* hb = h + (size_t)b * DEC;
  const float* wr = wW + (size_t)d * (ENC2 + DEC);
  float acc = wb[d];
  for (int f = 0; f < DEC; ++f) acc = fmaf(hb[f], wr[f], acc);
  hproj[idx] = acc;
}

// 3) pair-interleaved transpose: wWT[((k>>1)*512 + n)*2 + (k&1)] = wW[n][512+k]
//    -> a WMMA B-fragment {B[k][n], B[k+1][n]} is one aligned ds_load_b64.
__global__ void k_wwt(const float* __restrict__ wW, float* __restrict__ wWT) {
  int idx = blockIdx.x * blockDim.x + threadIdx.x;   // 1024*512
  int k = idx / DEC, n = idx % DEC;
  wWT[((size_t)(k >> 1) * DEC + n) * 2 + (k & 1)] =
      wW[(size_t)n * (ENC2 + DEC) + DEC + k];
}

// ---------------------------------------------------------------------------
// 4) energy GEMM, M=32 rows/block, double-buffered TDM staging,
//    fused tanh/vW reduction.  dynamic LDS:
//    Bt[2][64k*512n (interleaved pairs)] | At[2][32][66] | att_lds[32]
#define EN_CHUNK_B (64 * 512)
#define EN_CHUNK_A (32 * 66)
#define EN_SMEM_FLTS (2 * EN_CHUNK_B + 2 * EN_CHUNK_A + 32)
__global__ void __launch_bounds__(256)
k_energy(const float* __restrict__ enc, const float* __restrict__ wWT,
         const float* __restrict__ hproj, const float* __restrict__ vW,
         float* __restrict__ att) {
  extern __shared__ float smem[];
  // buffer pointers computed arithmetically (no pointer arrays of LDS consts)
  float* const Bt0     = smem;                             // 2 x EN_CHUNK_B
  float* const At0     = smem + 2 * EN_CHUNK_B;            // 2 x EN_CHUNK_A
  float* const att_lds = smem + 2 * EN_CHUNK_B + 2 * EN_CHUNK_A;  // [32]

  const int tid  = threadIdx.x;
  const int wave = tid >> 5;
  const int lane = tid & 31;
  const int nl   = lane & 15;
  const int hk   = (lane >> 4) << 1;

  const int rg = blockIdx.x;               // 4096 groups of 32 rows
  const int r0 = rg * 32;
  const int s  = r0 >> 6;
  const int b0 = r0 & 63;                  // 0 or 32
  const float* enc0 = enc + (size_t)r0 * ENC2;

  v8f acc[2][4];
#pragma unroll
  for (int mt = 0; mt < 2; ++mt)
#pragma unroll
    for (int j = 0; j < 4; ++j) acc[mt][j] = v8f{};

#if HAVE_TDM
  if (tid < 32) {   // wave 0 drives the Tensor Data Mover; prologue chunk 0
    tdm_load_2d((unsigned)(uintptr_t)Bt0, wWT, 32768u, 1u, 32768u, 1u,
                32768ull, 0u, 0u, 0u);
    tdm_load_2d((unsigned)(uintptr_t)At0, enc0, 64u, 32u, 64u, 32u,
                1024ull, 1u, 5u, 1u);
  }
#endif

  for (int c = 0; c < 16; ++c) {
    const int kc = c * 64;
#if HAVE_TDM
    if (tid < 32) {
      if (c + 1 < 16) {   // issue next chunk, then wait only for current one
        tdm_load_2d((unsigned)(uintptr_t)(Bt0 + ((c + 1) & 1) * EN_CHUNK_B),
                    wWT + (size_t)(kc + 64) * DEC, 32768u, 1u, 32768u, 1u,
                    32768ull, 0u, 0u, 0u);
        tdm_load_2d((unsigned)(uintptr_t)(At0 + ((c + 1) & 1) * EN_CHUNK_A),
                    enc0 + kc + 64, 64u, 32u, 64u, 32u, 1024ull, 1u, 5u, 1u);
        __builtin_amdgcn_s_wait_tensorcnt(2);
      } else {
        __builtin_amdgcn_s_wait_tensorcnt(0);
      }
    }
    __syncthreads();
#else
    {
      float* BT = Bt0 + (c & 1) * EN_CHUNK_B;
      float* AT = At0 + (c & 1) * EN_CHUNK_A;
      for (int i = tid; i < EN_CHUNK_B; i += 256) BT[i] = wWT[(size_t)kc * DEC + i];
      for (int i = tid; i < 32 * 64; i += 256) {
        int m = i >> 6, kk = i & 63;
        AT[m * 66 + kk] = enc0[(size_t)m * ENC2 + kc + kk];
      }
      __syncthreads();
    }
#endif
    const float* BT = Bt0 + (c & 1) * EN_CHUNK_B;
    const float* AT = At0 + (c & 1) * EN_CHUNK_A;
#pragma unroll
    for (int ks = 0; ks < 64; ks += 4) {
      const int kA = ks + hk;                       // even
      v2f a0 = *(const v2f*)&AT[nl * 66 + kA];      // ds b64
      v2f a1 = *(const v2f*)&AT[(16 + nl) * 66 + kA];
#pragma unroll
      for (int j = 0; j < 4; ++j) {
        const int n = wave * 64 + j * 16 + nl;
        v2f bf = *(const v2f*)&BT[((size_t)(kA >> 1) * DEC + n) * 2];  // ds b64
        acc[0][j] = __builtin_amdgcn_wmma_f32_16x16x4_f32(
            false, a0, false, bf, (short)0, acc[0][j], false, false);
        acc[1][j] = __builtin_amdgcn_wmma_f32_16x16x4_f32(
            false, a1, false, bf, (short)0, acc[1][j], false, false);
      }
    }
    __syncthreads();
  }

  if (tid < 32) att_lds[tid] = 0.0f;
  __syncthreads();
  const int mhi = (lane >= 16) ? 8 : 0;
#pragma unroll
  for (int mt = 0; mt < 2; ++mt) {
#pragma unroll
    for (int i = 0; i < 8; ++i) {
      const int m = mt * 16 + i + mhi;
      float t = 0.0f;
#pragma unroll
      for (int j = 0; j < 4; ++j) {
        const int n = wave * 64 + j * 16 + nl;
        t += vW[n] * tanhf(acc[mt][j][i] + hproj[(size_t)(b0 + m) * DEC + n]);
      }
      atomicAdd(&att_lds[m], t);                   // ds_add_f32
    }
  }
  __syncthreads();
  if (tid < 32) att[(size_t)(b0 + tid) * SRC + s] = att_lds[tid];
}

// ---------------------------------------------------------------------------
// 5) masked softmax; writes alignment into d_out
__global__ void __launch_bounds__(256)
k_softmax(const float* __restrict__ att, const int* __restrict__ mask,
          float* __restrict__ align) {
  __shared__ float red[256];
  const int b = blockIdx.x, tid = threadIdx.x;
  const float* ab = att + (size_t)b * SRC;
  const int* mb = mask + (size_t)b * SRC;

  float mx = -3.0e38f;
  for (int s = tid; s < SRC; s += 256) {
    float v = (mb[s] == 0) ? -1.0e10f : ab[s];
    mx = fmaxf(mx, v);
  }
  red[tid] = mx; __syncthreads();
  for (int o = 128; o > 0; o >>= 1) {
    if (tid < o) red[tid] = fmaxf(red[tid], red[tid + o]);
    __syncthreads();
  }
  mx = red[0]; __syncthreads();

  float sum = 0.0f;
  for (int s = tid; s < SRC; s += 256) {
    float v = (mb[s] == 0) ? -1.0e10f : ab[s];
    sum += __expf(v - mx);
  }
  red[tid] = sum; __syncthreads();
  for (int o = 128; o > 0; o >>= 1) {
    if (tid < o) red[tid] += red[tid + o];
    __syncthreads();
  }
  const float inv = 1.0f / red[0];
  for (int s = tid; s < SRC; s += 256) {
    float v = (mb[s] == 0) ? -1.0e10f : ab[s];
    align[(size_t)b * SRC + s] = __expf(v - mx) * inv;
  }
}

// 6) context[b][f] = sum_s align[b][s] * enc[s][b][f]   (b128 streaming)
__global__ void __launch_bounds__(256)
k_context(const float* __restrict__ enc, const float* __restrict__ align,
          float* __restrict__ ctx) {
  const int b = blockIdx.x, tid = threadIdx.x;
  float a0 = 0.f, a1 = 0.f, a2 = 0.f, a3 = 0.f;
  for (int s = 0; s < SRC; ++s) {
    const float al = align[(size_t)b * SRC + s];
    const float4 v = ((const float4*)(enc + ((size_t)s * BATCH + b) * ENC2))[tid];
    a0 = fmaf(al, v.x, a0); a1 = fmaf(al, v.y, a1);
    a2 = fmaf(al, v.z, a2); a3 = fmaf(al, v.w, a3);
  }
  float4* cb4 = (float4*)(ctx + (size_t)b * ENC2);
  cb4[tid] = make_float4(a0, a1, a2, a3);
}

// ---------------------------------------------------------------------------
// 7) single-step GRU (gate order r,z,n). rnn_in = [embedded | context]
__global__ void __launch_bounds__(256)
k_gru(const float* __restrict__ embedded, const float* __restrict__ ctx,
      const float* __restrict__ h, const float* __restrict__ Wih,
      const float* __restrict__ Whh, const float* __restrict__ bih,
      const float* __restrict__ bhh, float* __restrict__ h_new) {
  const int idx = blockIdx.x * blockDim.x + threadIdx.x;  // 64*512
  const int b = idx / DEC, d = idx % DEC;
  const int KI = ENC2 + EMB;  // 1280

  const float* wr = Wih + (size_t)d * KI;
  const float* wz = Wih + (size_t)(DEC + d) * KI;
  const float* wn = Wih + (size_t)(2 * DEC + d) * KI;
  float gxr = bih[d], gxz = bih[DEC + d], gxn = bih[2 * DEC + d];
  const float* eb = embedded + (size_t)b * EMB;
  for (int k = 0; k < EMB; ++k) {
    const float x = eb[k];
    gxr = fmaf(x, wr[k], gxr); gxz = fmaf(x, wz[k], gxz); gxn = fmaf(x, wn[k], gxn);
  }
  const float* cb = ctx + (size_t)b * ENC2;
  for (int k = 0; k < ENC2; ++k) {
    const float x = cb[k];
    gxr = fmaf(x, wr[EMB + k], gxr); gxz = fmaf(x, wz[EMB + k], gxz);
    gxn = fmaf(x, wn[EMB + k], gxn);
  }

  const float* ur = Whh + (size_t)d * DEC;
  const float* uz = Whh + (size_t)(DEC + d) * DEC;
  const float* un = Whh + (size_t)(2 * DEC + d) * DEC;
  float ghr = bhh[d], ghz = bhh[DEC + d], ghn = bhh[2 * DEC + d];
  const float* hb = h + (size_t)b * DEC;
  for (int k = 0; k < DEC; ++k) {
    const float x = hb[k];
    ghr = fmaf(x, ur[k], ghr); ghz = fmaf(x, uz[k], ghz); ghn = fmaf(x, un[k], ghn);
  }

  const float r = 1.0f / (1.0f + __expf(-(gxr + ghr)));
  const float z = 1.0f / (1.0f + __expf(-(gxz + ghz)));
  const float n = tanhf(gxn + r * ghn);
  h_new[idx] = (1.0f - z) * n + z * hb[d];
}

// 8) feat[b][0:1792] = [embedded(256) | h_new(512) | context(1024)]
__global__ void k_feat(const float* __restrict__ embedded, const float* __restrict__ h_new,
                       const float* __restrict__ ctx, float* __restrict__ feat) {
  const int idx = blockIdx.x * blockDim.x + threadIdx.x;  // 64*1792
  const int b = idx / 1792, k = idx % 1792;
  float v;
  if (k < EMB)            v = embedded[(size_t)b * EMB + k];
  else if (k < EMB + DEC) v = h_new[(size_t)b * DEC + (k - EMB)];
  else                    v = ctx[(size_t)b * ENC2 + (k - EMB - DEC)];
  feat[idx] = v;
}

// ---------------------------------------------------------------------------
// 9) prediction: predT = linW[M=vocab,K=1792] x featT[K,N=64], store transposed.
//    feat K-chunks double-buffer TDM-staged; linW streamed per-lane as b64.
__global__ void __launch_bounds__(256)
k_pred(const float* __restrict__ linW, const float* __restrict__ feat,
       const float* __restrict__ linb, float* __restrict__ pred) {
  __shared__ float Btile[2][64][66];   // feat chunks [b][kk], TDM-padded rows
  const int tid  = threadIdx.x;
  const int wave = tid >> 5;
  const int lane = tid & 31;
  const int nl   = lane & 15;
  const int hk   = (lane >> 4) << 1;
  const int KF   = ENC2 + DEC + EMB;   // 1792; 28 chunks of 64

  const int m0 = blockIdx.x * 128 + wave * 16;
  int arow = m0 + nl; if (arow >= VOCAB) arow = VOCAB - 1;
  const float* Arow = linW + (size_t)arow * KF;

  v8f acc[4] = {v8f{}, v8f{}, v8f{}, v8f{}};

#if HAVE_TDM
  if (tid < 32) {
    tdm_load_2d((unsigned)(uintptr_t)&Btile[0][0][0], feat, 64u, 64u, 64u, 64u,
                1792ull, 1u, 5u, 1u);
  }
#endif

  for (int c = 0; c < 28; ++c) {
    const int kc = c * 64;
#if HAVE_TDM
    if (tid < 32) {
      if (c + 1 < 28) {
        tdm_load_2d((unsigned)(uintptr_t)&Btile[(c + 1) & 1][0][0],
                    feat + kc + 64, 64u, 64u, 64u, 64u, 1792ull, 1u, 5u, 1u);
        __builtin_amdgcn_s_wait_tensorcnt(1);
      } else {
        __builtin_amdgcn_s_wait_tensorcnt(0);
      }
    }
    __syncthreads();
#else
    for (int i = tid; i < 64 * 64; i += 256) {
      int bb = i >> 6, kk = i & 63;
      Btile[c & 1][bb][kk] = feat[(size_t)bb * KF + kc + kk];
    }
    __syncthreads();
#endif
    if (kc + 64 < KF) __builtin_prefetch(Arow + kc + 64, 0, 1);
    const float (*BT)[66] = Btile[c & 1];
#pragma unroll
    for (int ks = 0; ks < 64; ks += 4) {
      const int kk = ks + hk;                      // even
      v2f a = *(const v2f*)(Arow + kc + kk);       // global b64, contiguous stream
#pragma unroll
      for (int j = 0; j < 4; ++j) {
        const int bb = j * 16 + nl;
        v2f bf = *(const v2f*)&BT[bb][kk];         // ds b64 (adjacent k,k+1)
        acc[j] = __builtin_amdgcn_wmma_f32_16x16x4_f32(
            false, a, false, bf, (short)0, acc[j], false, false);
      }
    }
    __syncthreads();
  }

  const int mhi = (lane >= 16) ? 8 : 0;
#pragma unroll
  for (int j = 0; j < 4; ++j) {
    const int bcol = j * 16 + nl;
#pragma unroll
    for (int i = 0; i < 8; ++i) {
      const int m = m0 + i + mhi;
      if (m < VOCAB) pred[(size_t)bcol * VOCAB + m] = acc[j][i] + linb[m];
    }
  }
}

// ---------------------------------------------------------------------------
extern "C" void kernel_launch(void* const* d_in, const int* in_sizes, int n_in,
                              void* d_out, int out_size, void* d_ws, size_t ws_size,
                              hipStream_t stream) {
  (void)in_sizes; (void)n_in; (void)out_size; (void)ws_size;
  const int*   input = (const int*)d_in[0];
  const float* h     = (const float*)d_in[1];
  const float* enc   = (const float*)d_in[2];
  const int*   mask  = (const int*)d_in[3];
  const float* emb   = (const float*)d_in[4];
  const float* wW    = (const float*)d_in[5];
  const float* wb    = (const float*)d_in[6];
  const float* vW    = (const float*)d_in[7];
  const float* Wih   = (const float*)d_in[8];
  const float* Whh   = (const float*)d_in[9];
  const float* bih   = (const float*)d_in[10];
  const float* bhh   = (const float*)d_in[11];
  const float* linW  = (const float*)d_in[12];
  const float* linb  = (const float*)d_in[13];

  float* out = (float*)d_out;
  float* ws  = (float*)d_ws;
  float* wWT      = ws + WS_WWT;
  float* hproj    = ws + WS_HPROJ;
  float* embedded = ws + WS_EMBD;
  float* att      = ws + WS_ATT;
  float* ctx      = ws + WS_CTX;
  float* feat     = ws + WS_FEAT;
  float* pred     = out + PRED_OFF;
  float* h_new    = out + HNEW_OFF;
  float* align    = out + ALIGN_OFF;

  k_embed<<<(BATCH * EMB) / 256, 256, 0, stream>>>(input, emb, embedded);
  k_hproj<<<(BATCH * DEC) / 256, 256, 0, stream>>>(h, wW, wb, hproj);
  k_wwt<<<(ENC2 * DEC) / 256, 256, 0, stream>>>(wW, wWT);

  k_energy<<<(SRC * BATCH) / 32, 256, EN_SMEM_FLTS * sizeof(float), stream>>>(
      enc, wWT, hproj, vW, att);
  k_softmax<<<BATCH, 256, 0, stream>>>(att, mask, align);
  k_context<<<BATCH, 256, 0, stream>>>(enc, align, ctx);

  k_gru<<<(BATCH * DEC) / 256, 256, 0, stream>>>(embedded, ctx, h, Wih, Whh, bih, bhh, h_new);
  k_feat<<<(BATCH * 1792) / 256, 256, 0, stream>>>(embedded, h_new, ctx, feat);

  k_pred<<<(VOCAB + 127) / 128, 256, 0, stream>>>(linW, feat, linb, pred);
}